// GAT_6717328851802
// MI455X (gfx1250) — compile-verified
//
#include <hip/hip_runtime.h>
#include <hip/hip_bf16.h>

#define IN_F 512
#define HIDC 64
#define HEADS 4
#define CLASSES 6
#define NEG_SLOPE 0.2f
#define HH 256  // HEADS*HIDC
#define LDS_STRIDE 514  // 512 + 2 floats padding (bank-conflict avoidance)

typedef float v2f __attribute__((ext_vector_type(2)));
typedef float v8f __attribute__((ext_vector_type(8)));

// ---------- helpers ----------
__device__ __forceinline__ int fkey(float f) {
  int i = __float_as_int(f);
  return i >= 0 ? i : (i ^ 0x7fffffff);  // order-preserving map float -> int
}
__device__ __forceinline__ float funkey(int k) {
  return __int_as_float(k >= 0 ? k : (k ^ 0x7fffffff));
}
__device__ __forceinline__ float lrelu(float v) { return v > 0.f ? v : NEG_SLOPE * v; }

// ---------- init: max-keys to INT_MIN ----------
__global__ void init_keys_kernel(int* mk1, int* mk2, int n) {
  int i = blockIdx.x * blockDim.x + threadIdx.x;
  if (i < 4 * n) mk1[i] = INT_MIN;
  if (i < n) mk2[i] = INT_MIN;
}

// ---------- GEMM1: h = x @ W1  (M=N_nodes, K=512, N=256) via fp32 WMMA ----------
__global__ __launch_bounds__(256) void gemm1_kernel(const float* __restrict__ x,
                                                    const float* __restrict__ W1,
                                                    float* __restrict__ h) {
  __shared__ float sA[16 * LDS_STRIDE];
  const int tid = threadIdx.x;
  const int rowbase = blockIdx.x * 16;

  // cooperative stage of 16x512 fp32 tile of x into LDS via async global->LDS DMA
  // (GLOBAL_LOAD_ASYNC_TO_LDS_B128, tracked by ASYNCcnt). Per lane: 16B moved
  // memory->LDS with no VGPR round-trip.
  for (int t = 0; t < 8; ++t) {
    int idx = tid + 256 * t;  // float4 index, 0..2047
    int r = idx >> 7;
    int c4 = idx & 127;
    const float* gsrc = x + (size_t)(rowbase + r) * IN_F + (size_t)c4 * 4;
    // low 32 bits of the flat shared-pointer == wave-relative LDS byte address
    unsigned laddr = (unsigned)(uintptr_t)(&sA[r * LDS_STRIDE + c4 * 4]);
    asm volatile("global_load_async_to_lds_b128 %0, %1, off"
                 :: "v"(laddr), "v"(gsrc)
                 : "memory");
  }
  asm volatile("s_wait_asynccnt 0x0" ::: "memory");
  __syncthreads();

  const int lane = tid & 31;
  const int wave = tid >> 5;
  const int m = lane & 15;
  const int kh = lane >> 4;  // 0: K=0,1  1: K=2,3
  const int colbase = wave * 32;

  v8f c0 = {};
  v8f c1 = {};
  const float* Wp = W1 + colbase + m;  // this lane's column (tile 0)

  for (int k = 0; k < IN_F; k += 4) {
    // A fragment: 16x4 fp32 -> 2 VGPRs (lane half selects K pair)
    v2f a = *(const v2f*)(&sA[m * LDS_STRIDE + k + 2 * kh]);
    int krow = k + 2 * kh;
    v2f b0, b1;
    b0.x = Wp[(size_t)krow * HH];
    b0.y = Wp[(size_t)(krow + 1) * HH];
    b1.x = Wp[(size_t)krow * HH + 16];
    b1.y = Wp[(size_t)(krow + 1) * HH + 16];
    if (k + 8 < IN_F) __builtin_prefetch(Wp + (size_t)(krow + 8) * HH, 0, 1);
    c0 = __builtin_amdgcn_wmma_f32_16x16x4_f32(false, a, false, b0, (short)0, c0, false, false);
    c1 = __builtin_amdgcn_wmma_f32_16x16x4_f32(false, a, false, b1, (short)0, c1, false, false);
  }

  // C/D layout: VGPR j, lanes<16 -> M=j, lanes>=16 -> M=j+8 ; N = lane&15
#pragma unroll
  for (int j = 0; j < 8; ++j) {
    int row = rowbase + j + 8 * kh;
    h[(size_t)row * HH + colbase + m] = c0[j];
    h[(size_t)row * HH + colbase + 16 + m] = c1[j];
  }
}

// ---------- per-node attention halves, layer 1 (wave per (node,head)) ----------
__global__ __launch_bounds__(256) void ecoef1_kernel(const float* __restrict__ h,
                                                     const float* __restrict__ a_src,
                                                     const float* __restrict__ a_dst,
                                                     float* __restrict__ esrc,
                                                     float* __restrict__ edst, int n_nodes) {
  int w = blockIdx.x * 8 + (threadIdx.x >> 5);
  int lane = threadIdx.x & 31;
  int n = w >> 2;
  int hd = w & 3;
  if (n >= n_nodes) return;
  const float* hp = h + (size_t)n * HH + hd * HIDC;
  const float* as = a_src + hd * HIDC;
  const float* ad = a_dst + hd * HIDC;
  float h0 = hp[lane], h1 = hp[lane + 32];
  float s = h0 * as[lane] + h1 * as[lane + 32];
  float d = h0 * ad[lane] + h1 * ad[lane + 32];
  for (int off = 16; off; off >>= 1) {
    s += __shfl_xor(s, off, 32);
    d += __shfl_xor(d, off, 32);
  }
  if (lane == 0) {
    esrc[n * 4 + hd] = s;
    edst[n * 4 + hd] = d;
  }
}

// ---------- layer-1 edge pass 1: segment max ----------
__global__ void edge1_max_kernel(const int* __restrict__ ei, const float* __restrict__ esrc,
                                 const float* __restrict__ edst, int* __restrict__ mkey,
                                 int nE, int nEP) {
  int t = blockIdx.x * blockDim.x + threadIdx.x;
  if (t >= nEP * 4) return;
  int e = t >> 2, hd = t & 3;
  int s, d;
  if (e < nE) { s = ei[e]; d = ei[nE + e]; } else { s = d = e - nE; }
  float l = lrelu(esrc[s * 4 + hd] + edst[d * 4 + hd]);
  atomicMax(&mkey[d * 4 + hd], fkey(l));
}

// ---------- layer-1 edge pass 2: segment sum of exp ----------
__global__ void edge1_sum_kernel(const int* __restrict__ ei, const float* __restrict__ esrc,
                                 const float* __restrict__ edst, const int* __restrict__ mkey,
                                 float* __restrict__ denom, int nE, int nEP) {
  int t = blockIdx.x * blockDim.x + threadIdx.x;
  if (t >= nEP * 4) return;
  int e = t >> 2, hd = t & 3;
  int s, d;
  if (e < nE) { s = ei[e]; d = ei[nE + e]; } else { s = d = e - nE; }
  float l = lrelu(esrc[s * 4 + hd] + edst[d * 4 + hd]);
  float p = __expf(l - funkey(mkey[d * 4 + hd]));
  atomicAdd(&denom[d * 4 + hd], p);
}

// ---------- layer-1 edge pass 3: weighted aggregation (wave per edge) ----------
__global__ __launch_bounds__(256) void edge1_agg_kernel(const int* __restrict__ ei,
                                                        const float* __restrict__ esrc,
                                                        const float* __restrict__ edst,
                                                        const int* __restrict__ mkey,
                                                        const float* __restrict__ denom,
                                                        const float* __restrict__ h,
                                                        float* __restrict__ agg, int nE, int nEP) {
  int w = blockIdx.x * 8 + (threadIdx.x >> 5);
  if (w >= nEP) return;
  int lane = threadIdx.x & 31;
  int s, d;
  if (w < nE) { s = ei[w]; d = ei[nE + w]; } else { s = d = w - nE; }
  float4 es = *(const float4*)(esrc + (size_t)s * 4);
  float4 ed = *(const float4*)(edst + (size_t)d * 4);
  int4 mk = *(const int4*)(mkey + (size_t)d * 4);
  float4 dn = *(const float4*)(denom + (size_t)d * 4);
  float alpha[4];
  { float l = lrelu(es.x + ed.x); alpha[0] = __expf(l - funkey(mk.x)) / dn.x; }
  { float l = lrelu(es.y + ed.y); alpha[1] = __expf(l - funkey(mk.y)) / dn.y; }
  { float l = lrelu(es.z + ed.z); alpha[2] = __expf(l - funkey(mk.z)) / dn.z; }
  { float l = lrelu(es.w + ed.w); alpha[3] = __expf(l - funkey(mk.w)) / dn.w; }
  const float* hs = h + (size_t)s * HH;
  float* ag = agg + (size_t)d * HH;
#pragma unroll
  for (int hd = 0; hd < 4; ++hd)
#pragma unroll
    for (int ii = 0; ii < 2; ++ii) {
      int idx = hd * HIDC + ii * 32 + lane;
      atomicAdd(&ag[idx], alpha[hd] * hs[idx]);
    }
}

// ---------- h2 = elu(agg1 + b1) ----------
__global__ void act_kernel(const float* __restrict__ agg, const float* __restrict__ b1,
                           float* __restrict__ h2, int total) {
  int i = blockIdx.x * blockDim.x + threadIdx.x;
  if (i >= total) return;
  float v = agg[i] + b1[i & (HH - 1)];
  h2[i] = v > 0.f ? v : (__expf(v) - 1.0f);
}

// ---------- layer-2 node work: o2 = h2 @ W2 ; e_src2/e_dst2 (wave per node) ----------
__global__ __launch_bounds__(256) void layer2_node_kernel(const float* __restrict__ h2,
                                                          const float* __restrict__ W2,
                                                          const float* __restrict__ a_src2,
                                                          const float* __restrict__ a_dst2,
                                                          float* __restrict__ o2,
                                                          float* __restrict__ esrc2,
                                                          float* __restrict__ edst2, int n_nodes) {
  int n = blockIdx.x * 8 + (threadIdx.x >> 5);
  int lane = threadIdx.x & 31;
  if (n >= n_nodes) return;
  const float* hp = h2 + (size_t)n * HH;
  float acc[CLASSES] = {0.f, 0.f, 0.f, 0.f, 0.f, 0.f};
  for (int j = 0; j < 8; ++j) {
    int k = lane + 32 * j;
    float hv = hp[k];
    const float* wr = W2 + (size_t)k * CLASSES;
#pragma unroll
    for (int c = 0; c < CLASSES; ++c) acc[c] += hv * wr[c];
  }
#pragma unroll
  for (int c = 0; c < CLASSES; ++c)
    for (int off = 16; off; off >>= 1) acc[c] += __shfl_xor(acc[c], off, 32);
  if (lane == 0) {
    float s = 0.f, d = 0.f;
#pragma unroll
    for (int c = 0; c < CLASSES; ++c) {
      o2[(size_t)n * CLASSES + c] = acc[c];
      s += acc[c] * a_src2[c];
      d += acc[c] * a_dst2[c];
    }
    esrc2[n] = s;
    edst2[n] = d;
  }
}

// ---------- layer-2 edge passes (thread per edge) ----------
__global__ void edge2_max_kernel(const int* __restrict__ ei, const float* __restrict__ es,
                                 const float* __restrict__ ed, int* __restrict__ mkey,
                                 int nE, int nEP) {
  int e = blockIdx.x * blockDim.x + threadIdx.x;
  if (e >= nEP) return;
  int s, d;
  if (e < nE) { s = ei[e]; d = ei[nE + e]; } else { s = d = e - nE; }
  atomicMax(&mkey[d], fkey(lrelu(es[s] + ed[d])));
}

__global__ void edge2_sum_kernel(const int* __restrict__ ei, const float* __restrict__ es,
                                 const float* __restrict__ ed, const int* __restrict__ mkey,
                                 float* __restrict__ denom, int nE, int nEP) {
  int e = blockIdx.x * blockDim.x + threadIdx.x;
  if (e >= nEP) return;
  int s, d;
  if (e < nE) { s = ei[e]; d = ei[nE + e]; } else { s = d = e - nE; }
  float p = __expf(lrelu(es[s] + ed[d]) - funkey(mkey[d]));
  atomicAdd(&denom[d], p);
}

__global__ void edge2_agg_kernel(const int* __restrict__ ei, const float* __restrict__ es,
                                 const float* __restrict__ ed, const int* __restrict__ mkey,
                                 const float* __restrict__ denom, const float* __restrict__ o2,
                                 float* __restrict__ agg2, int nE, int nEP) {
  int e = blockIdx.x * blockDim.x + threadIdx.x;
  if (e >= nEP) return;
  int s, d;
  if (e < nE) { s = ei[e]; d = ei[nE + e]; } else { s = d = e - nE; }
  float alpha = __expf(lrelu(es[s] + ed[d]) - funkey(mkey[d])) / denom[d];
#pragma unroll
  for (int c = 0; c < CLASSES; ++c)
    atomicAdd(&agg2[(size_t)d * CLASSES + c], alpha * o2[(size_t)s * CLASSES + c]);
}

// ---------- final: log_softmax(agg2 + b2) ----------
__global__ void final_kernel(const float* __restrict__ agg2, const float* __restrict__ b2,
                             float* __restrict__ out, int n_nodes) {
  int n = blockIdx.x * blockDim.x + threadIdx.x;
  if (n >= n_nodes) return;
  float v[CLASSES];
  float mx = -INFINITY;
#pragma unroll
  for (int c = 0; c < CLASSES; ++c) {
    v[c] = agg2[(size_t)n * CLASSES + c] + b2[c];
    mx = fmaxf(mx, v[c]);
  }
  float ssum = 0.f;
#pragma unroll
  for (int c = 0; c < CLASSES; ++c) ssum += __expf(v[c] - mx);
  float lse = mx + __logf(ssum);
#pragma unroll
  for (int c = 0; c < CLASSES; ++c) out[(size_t)n * CLASSES + c] = v[c] - lse;
}

// ---------- host launch ----------
extern "C" void kernel_launch(void* const* d_in, const int* in_sizes, int n_in,
                              void* d_out, int out_size, void* d_ws, size_t ws_size,
                              hipStream_t stream) {
  const float* x = (const float*)d_in[0];
  const int* ei = (const int*)d_in[1];
  const float* W1 = (const float*)d_in[2];
  const float* a_src1 = (const float*)d_in[3];
  const float* a_dst1 = (const float*)d_in[4];
  const float* b1 = (const float*)d_in[5];
  const float* W2 = (const float*)d_in[6];
  const float* a_src2 = (const float*)d_in[7];
  const float* a_dst2 = (const float*)d_in[8];
  const float* b2 = (const float*)d_in[9];
  float* out = (float*)d_out;

  const int nN = in_sizes[0] / IN_F;   // 50000
  const int nE = in_sizes[1] / 2;      // 800000
  const int nEP = nE + nN;             // edges + self loops

  float* ws = (float*)d_ws;
  size_t off = 0;
  float* hfeat = ws + off; off += (size_t)nN * HH;      // also reused as h2
  float* agg1  = ws + off; off += (size_t)nN * HH;
  float* esrc1 = ws + off; off += (size_t)nN * 4;
  float* edst1 = ws + off; off += (size_t)nN * 4;
  int*   mkey1 = (int*)(ws + off); off += (size_t)nN * 4;
  float* den1  = ws + off; off += (size_t)nN * 4;
  float* o2    = ws + off; off += (size_t)nN * CLASSES;
  float* esrc2 = ws + off; off += (size_t)nN;
  float* edst2 = ws + off; off += (size_t)nN;
  int*   mkey2 = (int*)(ws + off); off += (size_t)nN;
  float* den2  = ws + off; off += (size_t)nN;
  float* agg2  = ws + off; off += (size_t)nN * CLASSES;

  // per-call state init (graph-capture safe)
  hipMemsetAsync(agg1, 0, (size_t)nN * HH * sizeof(float), stream);
  hipMemsetAsync(den1, 0, (size_t)nN * 4 * sizeof(float), stream);
  hipMemsetAsync(agg2, 0, (size_t)nN * CLASSES * sizeof(float), stream);
  hipMemsetAsync(den2, 0, (size_t)nN * sizeof(float), stream);
  init_keys_kernel<<<(nN * 4 + 255) / 256, 256, 0, stream>>>(mkey1, mkey2, nN);

  // layer 1
  gemm1_kernel<<<nN / 16, 256, 0, stream>>>(x, W1, hfeat);
  ecoef1_kernel<<<(nN * 4 + 7) / 8, 256, 0, stream>>>(hfeat, a_src1, a_dst1, esrc1, edst1, nN);
  edge1_max_kernel<<<((size_t)nEP * 4 + 255) / 256, 256, 0, stream>>>(ei, esrc1, edst1, mkey1, nE, nEP);
  edge1_sum_kernel<<<((size_t)nEP * 4 + 255) / 256, 256, 0, stream>>>(ei, esrc1, edst1, mkey1, den1, nE, nEP);
  edge1_agg_kernel<<<(nEP + 7) / 8, 256, 0, stream>>>(ei, esrc1, edst1, mkey1, den1, hfeat, agg1, nE, nEP);
  act_kernel<<<((size_t)nN * HH + 255) / 256, 256, 0, stream>>>(agg1, b1, hfeat, nN * HH);

  // layer 2
  layer2_node_kernel<<<(nN + 7) / 8, 256, 0, stream>>>(hfeat, W2, a_src2, a_dst2, o2, esrc2, edst2, nN);
  edge2_max_kernel<<<(nEP + 255) / 256, 256, 0, stream>>>(ei, esrc2, edst2, mkey2, nE, nEP);
  edge2_sum_kernel<<<(nEP + 255) / 256, 256, 0, stream>>>(ei, esrc2, edst2, mkey2, den2, nE, nEP);
  edge2_agg_kernel<<<(nEP + 255) / 256, 256, 0, stream>>>(ei, esrc2, edst2, mkey2, den2, o2, agg2, nE, nEP);

  // output
  final_kernel<<<(nN + 255) / 256, 256, 0, stream>>>(agg2, b2, out, nN);
}